// SimpleGCN_50276887167484
// MI455X (gfx1250) — compile-verified
//
#include <hip/hip_runtime.h>

typedef float v2f __attribute__((ext_vector_type(2)));
typedef float v8f __attribute__((ext_vector_type(8)));

#define IN_CH 128
#define HID_CH 128
#define N_CLASSES 64

// ---------------------------------------------------------------------------
// Elementwise / init kernels
// ---------------------------------------------------------------------------
__global__ void zero_kernel(float* __restrict__ p, int n) {
  int i = blockIdx.x * blockDim.x + threadIdx.x;
  int stride = gridDim.x * blockDim.x;
  for (; i < n; i += stride) p[i] = 0.0f;
}

__global__ void deg_kernel(const int* __restrict__ dst, float* __restrict__ deg, int E) {
  int i = blockIdx.x * blockDim.x + threadIdx.x;
  int stride = gridDim.x * blockDim.x;
  for (; i < E; i += stride) unsafeAtomicAdd(&deg[dst[i]], 1.0f);
}

// dinv = rsqrt(deg + 1)  (in place over deg buffer)
__global__ void dinv_kernel(float* __restrict__ deg, int n) {
  int i = blockIdx.x * blockDim.x + threadIdx.x;
  int stride = gridDim.x * blockDim.x;
  for (; i < n; i += stride) deg[i] = rsqrtf(deg[i] + 1.0f);
}

// z[n,c] = relu(agg[n,c] + bias[c])    (C power of two, cmask = C-1)
__global__ void bias_relu_kernel(const float* __restrict__ agg,
                                 const float* __restrict__ bias,
                                 float* __restrict__ z,
                                 int total, int cmask) {
  int i = blockIdx.x * blockDim.x + threadIdx.x;
  int stride = gridDim.x * blockDim.x;
  for (; i < total; i += stride) z[i] = fmaxf(agg[i] + bias[i & cmask], 0.0f);
}

// out[n,c] += bias[c]
__global__ void bias_add_kernel(float* __restrict__ out,
                                const float* __restrict__ bias,
                                int total, int cmask) {
  int i = blockIdx.x * blockDim.x + threadIdx.x;
  int stride = gridDim.x * blockDim.x;
  for (; i < total; i += stride) out[i] = out[i] + bias[i & cmask];
}

// ---------------------------------------------------------------------------
// Edge scatter: one wave32 per edge. agg[dst] += h[src] * dinv[src]*dinv[dst]
// Each lane owns C/32 contiguous channels: one b128/b64 load per lane, then
// native global_atomic_add_f32 into L2 (192MB L2 holds the whole working set).
// ---------------------------------------------------------------------------
template <int C>
__global__ void edge_scatter_kernel(const float* __restrict__ h,
                                    const int* __restrict__ src,
                                    const int* __restrict__ dst,
                                    const float* __restrict__ dinv,
                                    float* __restrict__ agg,
                                    int E) {
  constexpr int VEC = C / 32;  // 4 (C=128) or 2 (C=64)
  int wid = (blockIdx.x * blockDim.x + threadIdx.x) >> 5;  // edge id
  int lane = threadIdx.x & 31;
  if (wid >= E) return;
  int s = src[wid];
  int d = dst[wid];
  float norm = dinv[s] * dinv[d];
  const float* hrow = h + (size_t)s * C + lane * VEC;
  float* arow = agg + (size_t)d * C + lane * VEC;
  if constexpr (VEC == 4) {
    float4 v = *(const float4*)hrow;  // global_load_b128
    unsafeAtomicAdd(&arow[0], v.x * norm);
    unsafeAtomicAdd(&arow[1], v.y * norm);
    unsafeAtomicAdd(&arow[2], v.z * norm);
    unsafeAtomicAdd(&arow[3], v.w * norm);
  } else {
    float2 v = *(const float2*)hrow;  // global_load_b64
    unsafeAtomicAdd(&arow[0], v.x * norm);
    unsafeAtomicAdd(&arow[1], v.y * norm);
  }
}

// ---------------------------------------------------------------------------
// FP32 WMMA GEMM with fused self-loop epilogue:
//   D[M,N]   = A[M,K] @ B[K,N]          (row-major)
//   agg[M,N] = D * dinv[row]^2          (initializes the aggregation buffer)
// One wave computes one 16x16 tile of D via V_WMMA_F32_16X16X4_F32, K step 4.
// Requires M%16==0, N%16==0, K%4==0 (all hold: 50000/128/64).
//
// Fragment layouts (ISA 7.12.2, wave32):
//   A 16x4 f32 : lanes 0-15 -> rows M=lane, (x,y) = K=0,1
//                lanes 16-31 -> rows M=lane-16, (x,y) = K=2,3
//   B 4x16 f32 : mirrored K split: lanes 0-15 (x,y) = rows K=0,1, col N=lane
//                lanes 16-31 (x,y) = rows K=2,3, col N=lane-16
//   C/D 16x16  : element i: lanes 0-15 -> (M=i, N=lane),
//                           lanes 16-31 -> (M=8+i, N=lane-16)
// ---------------------------------------------------------------------------
__global__ void gemm_wmma_f32_kernel(const float* __restrict__ A,
                                     const float* __restrict__ B,
                                     float* __restrict__ D,
                                     const float* __restrict__ dinv,
                                     float* __restrict__ agg,
                                     int M, int K, int N) {
  int wid = (blockIdx.x * blockDim.x + threadIdx.x) >> 5;
  int lane = threadIdx.x & 31;
  int tilesN = N >> 4;
  int totalTiles = (M >> 4) * tilesN;
  if (wid >= totalTiles) return;  // wave-uniform: EXEC all-ones inside

  int tm = wid / tilesN;
  int tn = wid - tm * tilesN;
  int row0 = tm << 4;
  int col0 = tn << 4;

  int lr = lane & 15;            // 0..15: row (A) / col (B)
  int khalf = (lane >> 4) << 1;  // 0 for lanes 0-15, 2 for lanes 16-31

  const float* arow = A + (size_t)(row0 + lr) * K;
  v8f acc = {};

  for (int k0 = 0; k0 < K; k0 += 4) {
    v2f a, b;
    a.x = arow[k0 + khalf];
    a.y = arow[k0 + khalf + 1];
    b.x = B[(size_t)(k0 + khalf) * N + col0 + lr];
    b.y = B[(size_t)(k0 + khalf + 1) * N + col0 + lr];
    acc = __builtin_amdgcn_wmma_f32_16x16x4_f32(
        /*neg_a=*/false, a, /*neg_b=*/false, b,
        /*c_mod=*/(short)0, acc, /*reuse_a=*/false, /*reuse_b=*/false);
  }

  int mbase = row0 + ((lane >> 4) << 3);  // +8 for upper half-lanes
  int col = col0 + lr;
#pragma unroll
  for (int i = 0; i < 8; ++i) {
    int row = mbase + i;
    float v = acc[i];
    D[(size_t)row * N + col] = v;
    float dv = dinv[row];
    agg[(size_t)row * N + col] = v * dv * dv;  // self-loop contribution
  }
}

// ---------------------------------------------------------------------------
// Launch
// ---------------------------------------------------------------------------
extern "C" void kernel_launch(void* const* d_in, const int* in_sizes, int n_in,
                              void* d_out, int out_size, void* d_ws, size_t ws_size,
                              hipStream_t stream) {
  const float* x  = (const float*)d_in[0];   // [N, 128]
  const int* ei   = (const int*)d_in[1];     // [2, E]
  const float* W1 = (const float*)d_in[2];   // [128, 128]
  const float* b1 = (const float*)d_in[3];   // [128]
  const float* W2 = (const float*)d_in[4];   // [128, 64]
  const float* b2 = (const float*)d_in[5];   // [64]
  float* out = (float*)d_out;                // [N, 64]

  const int n_nodes = in_sizes[0] / IN_CH;   // 50000
  const int E = in_sizes[1] / 2;             // 640000
  const int* src = ei;
  const int* dst = ei + E;

  // Workspace layout (floats)
  float* ws = (float*)d_ws;
  float* dinv = ws;                               // [n_nodes]
  float* h1   = dinv + n_nodes;                   // [n_nodes, 128]
  float* agg1 = h1 + (size_t)n_nodes * HID_CH;    // [n_nodes, 128]
  float* h2   = agg1 + (size_t)n_nodes * HID_CH;  // [n_nodes, 64]
  // z1 reuses h1's buffer after scatter1 consumes h1
  (void)ws_size; (void)n_in; (void)out_size;

  const int TB = 256;
  const int totalH = n_nodes * HID_CH;     // 6.4M
  const int totalC = n_nodes * N_CLASSES;  // 3.2M

  // 1) degree -> dinv (in place)
  zero_kernel<<<256, TB, 0, stream>>>(dinv, n_nodes);
  deg_kernel<<<(E + TB - 1) / TB, TB, 0, stream>>>(dst, dinv, E);
  dinv_kernel<<<256, TB, 0, stream>>>(dinv, n_nodes);

  // 2) h1 = x @ W1 (fp32 WMMA), fused agg1 = h1 * dinv^2
  {
    int tiles = (n_nodes / 16) * (HID_CH / 16);
    int blocks = (tiles * 32 + TB - 1) / TB;
    gemm_wmma_f32_kernel<<<blocks, TB, 0, stream>>>(x, W1, h1, dinv, agg1,
                                                    n_nodes, IN_CH, HID_CH);
  }

  // 3) scatter edges into agg1 ; z1 = relu(agg1 + b1) -> h1 buffer
  edge_scatter_kernel<HID_CH><<<(E * 32 + TB - 1) / TB, TB, 0, stream>>>(
      h1, src, dst, dinv, agg1, E);
  bias_relu_kernel<<<4096, TB, 0, stream>>>(agg1, b1, h1, totalH, HID_CH - 1);

  // 4) h2 = z1 @ W2 (fp32 WMMA), fused out = h2 * dinv^2
  {
    int tiles = (n_nodes / 16) * (N_CLASSES / 16);
    int blocks = (tiles * 32 + TB - 1) / TB;
    gemm_wmma_f32_kernel<<<blocks, TB, 0, stream>>>(h1, W2, h2, dinv, out,
                                                    n_nodes, HID_CH, N_CLASSES);
  }

  // 5) scatter edges into out ; out += b2
  edge_scatter_kernel<N_CLASSES><<<(E * 32 + TB - 1) / TB, TB, 0, stream>>>(
      h2, src, dst, dinv, out, E);
  bias_add_kernel<<<4096, TB, 0, stream>>>(out, b2, totalC, N_CLASSES - 1);
}